// GraphConvLayer_4879082848618
// MI455X (gfx1250) — compile-verified
//
#include <hip/hip_runtime.h>
#include <hip/hip_bf16.h>
#include <math.h>

typedef float v2f __attribute__((ext_vector_type(2)));
typedef float v4f __attribute__((ext_vector_type(4)));
typedef float v8f __attribute__((ext_vector_type(8)));

#define FDIM 64

// ---------------- zero workspace ----------------
__global__ void gc_zero_kernel(float* __restrict__ p, long long n) {
    long long i = (long long)blockIdx.x * blockDim.x + threadIdx.x;
    long long stride = (long long)gridDim.x * blockDim.x;
    for (; i < n; i += stride) p[i] = 0.0f;
}

// ---------------- degree: deg[col[e]] += 1 ----------------
__global__ void gc_degree_kernel(const int* __restrict__ col,
                                 float* __restrict__ deg, int E) {
    int e = blockIdx.x * blockDim.x + threadIdx.x;
    if (e < E) atomicAdd(&deg[col[e]], 1.0f);
}

// ---------------- scatter: agg[col] += value * x[row] ----------------
// one thread per (edge, feature); x and agg are L2-resident (12.8 MB each
// vs 192 MB L2), so gather + float atomics stay on-chip.
__global__ void gc_scatter_kernel(const float* __restrict__ x,
                                  const int* __restrict__ row,
                                  const int* __restrict__ col,
                                  const float* __restrict__ deg,
                                  float* __restrict__ agg, int E) {
    long long t = (long long)blockIdx.x * blockDim.x + threadIdx.x;
    int e = (int)(t >> 6);
    int f = (int)(t & 63);
    if (e >= E) return;
    int r = row[e];
    int c = col[e];
    float dr = deg[r];
    float dc = deg[c];
    if (dr > 0.0f && dc > 0.0f) {
        float v = (1.0f / sqrtf(dr)) * (1.0f / sqrtf(dc));
        atomicAdd(&agg[(size_t)c * FDIM + f], v * x[(size_t)r * FDIM + f]);
    }
}

// ---------------- out = agg @ W^T + b via V_WMMA_F32_16X16X4_F32 ----------------
// Per wave: one 16-row tile of agg, all 4 column tiles (64 outputs).
// A layout: lane<16 -> {A[m][4k],A[m][4k+1]}, lane>=16 -> {A[m][4k+2],A[m][4k+3]}
// B layout: lane&15 = output column n; same K-half split; B[k][n] = W[n][k]
// C/D layout: VGPR j -> rows j / j+8 (by lane half), column = lane&15
__global__ __launch_bounds__(256) void gc_gemm_wmma_kernel(
    const float* __restrict__ agg, const float* __restrict__ W,
    const float* __restrict__ bias, float* __restrict__ out, int n) {
    __shared__ float sW[FDIM * FDIM];        // W row-major [o][k], 16 KB
    __shared__ float sA[8][16 * FDIM];       // per-wave 16x64 tile, 32 KB

    const int tid  = threadIdx.x;
    const int wave = tid >> 5;
    const int lane = tid & 31;
    const int m    = lane & 15;              // A row within tile / B,D column
    const int koff = (lane >> 4) << 1;       // 0 or 2 (K-half)
    const int r0   = (blockIdx.x * 8 + wave) * 16;

    // stage W (4096 floats) cooperatively: 256 threads x 4 x float4
    {
        const v4f* Wv = (const v4f*)W;
        v4f* sWv = (v4f*)sW;
#pragma unroll
        for (int i = 0; i < 4; ++i) sWv[i * 256 + tid] = Wv[i * 256 + tid];
    }
    // stage this wave's 16x64 agg tile: 256 float4, 32 lanes x 8
    {
        v4f* dst = (v4f*)sA[wave];
#pragma unroll
        for (int i = 0; i < 8; ++i) {
            int idx = i * 32 + lane;         // float4 index in tile [0,256)
            int r = r0 + (idx >> 4);         // 16 float4 per row
            v4f v = {0.0f, 0.0f, 0.0f, 0.0f};
            if (r < n) v = ((const v4f*)(agg + (size_t)r * FDIM))[idx & 15];
            dst[idx] = v;
        }
    }
    __syncthreads();

    if (r0 >= n) return;                      // wave-uniform: EXEC stays all-1s

    const float* Arow = &sA[wave][m * FDIM];

    v8f acc[4];
#pragma unroll
    for (int t = 0; t < 4; ++t) {
        float bv = bias[t * 16 + m];          // seed C with bias
#pragma unroll
        for (int j = 0; j < 8; ++j) acc[t][j] = bv;
    }

#pragma unroll
    for (int k = 0; k < 16; ++k) {
        int kk = k * 4 + koff;
        v2f a = *(const v2f*)(Arow + kk);
#pragma unroll
        for (int t = 0; t < 4; ++t) {
            v2f b = *(const v2f*)(sW + (t * 16 + m) * FDIM + kk);
            acc[t] = __builtin_amdgcn_wmma_f32_16x16x4_f32(
                /*neg_a=*/false, a, /*neg_b=*/false, b,
                /*c_mod=*/(short)0, acc[t],
                /*reuse_a=*/false, /*reuse_b=*/false);
        }
    }

    const int rbase = r0 + ((lane >> 4) << 3);   // +8 for upper lane half
#pragma unroll
    for (int t = 0; t < 4; ++t) {
#pragma unroll
        for (int j = 0; j < 8; ++j) {
            int r = rbase + j;
            if (r < n) out[(size_t)r * FDIM + t * 16 + m] = acc[t][j];
        }
    }
}

extern "C" void kernel_launch(void* const* d_in, const int* in_sizes, int n_in,
                              void* d_out, int out_size, void* d_ws, size_t ws_size,
                              hipStream_t stream) {
    const float* x  = (const float*)d_in[0];   // [N, 64]
    // d_in[1] = x0 : unused (use_init=False)
    const int*   ei = (const int*)d_in[2];     // [2, E] flattened
    const float* Ww = (const float*)d_in[3];   // [64, 64]
    const float* Wb = (const float*)d_in[4];   // [64]
    float* out = (float*)d_out;

    const int N = in_sizes[0] / FDIM;
    const int E = in_sizes[2] / 2;
    const int* rowp = ei;
    const int* colp = ei + E;

    // workspace layout: deg [N floats] | agg [N*64 floats] (64-float aligned)
    float* deg = (float*)d_ws;
    const long long aggOff = ((long long)N + 63) & ~63LL;
    float* agg = deg + aggOff;

    // 1) zero deg + agg
    const long long zn = aggOff + (long long)N * FDIM;
    gc_zero_kernel<<<2048, 256, 0, stream>>>((float*)d_ws, zn);

    // 2) degrees
    gc_degree_kernel<<<(E + 255) / 256, 256, 0, stream>>>(colp, deg, E);

    // 3) edge-parallel scatter (E*64 threads)
    const long long st = (long long)E * FDIM;
    gc_scatter_kernel<<<(int)((st + 255) / 256), 256, 0, stream>>>(
        x, rowp, colp, deg, agg, E);

    // 4) WMMA GEMM + bias (128 rows per block)
    gc_gemm_wmma_kernel<<<(N + 127) / 128, 256, 0, stream>>>(agg, Ww, Wb, out, N);
}